// ConvDeepSet_463856468356
// MI455X (gfx1250) — compile-verified
//
#include <hip/hip_runtime.h>

// ConvDeepSet on MI455X (gfx1250, wave32).
// Phase A (trans-bound): per-m Gaussian aggregation with v_exp_f32, LDS-staged x/y.
// Phase B (WMMA): feats(16x8) x w^T(8x16) projection via V_WMMA_F32_16X16X4_F32.

#define B_      16
#define N_      1024
#define M_      1024
#define CIN_    7
#define C_      8
#define COUT_   16
#define TILE_M  128
#define NTILES  (M_ / TILE_M)   // 8
#define THREADS 128             // 4 waves of 32
#define EPS_    1e-8f

typedef __attribute__((ext_vector_type(2))) float v2f;
typedef __attribute__((ext_vector_type(8))) float v8f;

__global__ __launch_bounds__(THREADS)
void convdeepset_kernel(const float* __restrict__ x,
                        const float* __restrict__ y,
                        const float* __restrict__ t,
                        const float* __restrict__ sigma,
                        const float* __restrict__ w,
                        const float* __restrict__ bias,
                        float* __restrict__ out)
{
    __shared__ float sx[N_];            //  4 KB : x[b, :]
    __shared__ float sy[N_][C_];        // 32 KB : y_cat[b, :, :]  (c==0 -> 1.0)
    __shared__ float st[TILE_M];        //        t[b, m-tile]
    __shared__ float scoef[C_];         //        -0.5 * exp(-2*sigma_c) * log2(e)
    __shared__ float sfeat[TILE_M][9];  //  4.5KB: feats tile, stride 9 (bank-conflict free)

    const int tid = threadIdx.x;
    const int b   = blockIdx.x / NTILES;
    const int mt  = blockIdx.x % NTILES;
    const int m0  = mt * TILE_M;

    // ---- stage inputs into LDS -------------------------------------------
    for (int i = tid; i < N_; i += THREADS)
        sx[i] = x[b * N_ + i];

    for (int i = tid; i < N_ * C_; i += THREADS) {
        const int n = i >> 3;           // / C_
        const int c = i & 7;            // % C_
        sy[n][c] = (c == 0) ? 1.0f : y[(b * N_ + n) * CIN_ + (c - 1)];
    }

    st[tid] = t[b * M_ + m0 + tid];     // TILE_M == THREADS

    if (tid < C_) {
        const float LOG2E = 1.4426950408889634f;
        // scales = exp(sigma)  =>  1/scales^2 = exp(-2*sigma) = exp2(-2*sigma*log2e)
        const float inv_s2 = __builtin_amdgcn_exp2f(-2.0f * sigma[tid] * LOG2E);
        scoef[tid] = -0.5f * inv_s2 * LOG2E;   // so wt = exp2(d * coef[c])
    }
    __syncthreads();

    // ---- Phase A: Gaussian aggregation (one thread per output m) ---------
    float coef[C_];
    #pragma unroll
    for (int c = 0; c < C_; ++c) coef[c] = scoef[c];

    float acc[C_];
    #pragma unroll
    for (int c = 0; c < C_; ++c) acc[c] = 0.0f;

    const float tm = st[tid];

    #pragma unroll 2
    for (int n = 0; n < N_; ++n) {
        const float dx = sx[n] - tm;    // LDS broadcast read
        const float d  = dx * dx;
        #pragma unroll
        for (int c = 0; c < C_; ++c)
            acc[c] += sy[n][c] * __builtin_amdgcn_exp2f(d * coef[c]);
    }

    // ---- normalize -> feats tile in LDS ----------------------------------
    const float density = acc[0];
    const float inv_den = 1.0f / (density + EPS_);
    sfeat[tid][0] = density;
    #pragma unroll
    for (int c = 1; c < C_; ++c)
        sfeat[tid][c] = acc[c] * inv_den;
    __syncthreads();

    // ---- Phase B: projection via V_WMMA_F32_16X16X4_F32 ------------------
    // out[m, o] = sum_c feats[m, c] * w[o, c] + bias[o]
    // A (16x4 f32, 2 VGPRs): lanes 0-15 rows M, .x=K0/.y=K1 ; lanes 16-31 .x=K2/.y=K3
    // B (4x16 f32, 2 VGPRs): lanes 0-15 cols N, .x=K0/.y=K1 ; lanes 16-31 .x=K2/.y=K3
    // D (16x16 f32, 8 VGPRs): VGPR v -> row v (lanes 0-15) / row v+8 (lanes 16-31)
    const int wave = tid >> 5;
    const int lane = tid & 31;
    const int half = lane >> 4;         // 0: lanes 0-15, 1: lanes 16-31
    const int l16  = lane & 15;

    // B fragments from w[o*C + c]; B[k][n] = w[n][k]
    v2f bf0, bf1;
    bf0.x = w[l16 * C_ + (2 * half + 0)];
    bf0.y = w[l16 * C_ + (2 * half + 1)];
    bf1.x = w[l16 * C_ + (2 * half + 4)];
    bf1.y = w[l16 * C_ + (2 * half + 5)];
    const float bo = bias[l16];

    #pragma unroll
    for (int s = 0; s < 2; ++s) {       // each wave handles 2 sub-tiles of 16 m
        const int msub = (wave * 2 + s) * 16;
        const int mrow = msub + l16;

        v2f a0, a1;
        a0.x = sfeat[mrow][2 * half + 0];
        a0.y = sfeat[mrow][2 * half + 1];
        a1.x = sfeat[mrow][2 * half + 4];
        a1.y = sfeat[mrow][2 * half + 5];

        v8f d8 = {};
        d8 = __builtin_amdgcn_wmma_f32_16x16x4_f32(
                 false, a0, false, bf0, (short)0, d8, false, false);
        d8 = __builtin_amdgcn_wmma_f32_16x16x4_f32(
                 false, a1, false, bf1, (short)0, d8, false, false);

        #pragma unroll
        for (int v = 0; v < 8; ++v) {
            const int m = m0 + msub + v + half * 8;
            out[(b * M_ + m) * COUT_ + l16] = d8[v] + bo;
        }
    }
}

extern "C" void kernel_launch(void* const* d_in, const int* in_sizes, int n_in,
                              void* d_out, int out_size, void* d_ws, size_t ws_size,
                              hipStream_t stream)
{
    const float* x     = (const float*)d_in[0];   // (16, 1024, 1)
    const float* y     = (const float*)d_in[1];   // (16, 1024, 7)
    const float* t     = (const float*)d_in[2];   // (16, 1024, 1)
    const float* sigma = (const float*)d_in[3];   // (8,)
    const float* w     = (const float*)d_in[4];   // (16, 8)
    const float* bias  = (const float*)d_in[5];   // (16,)
    float* out         = (float*)d_out;           // (16, 1024, 16)

    (void)in_sizes; (void)n_in; (void)out_size; (void)d_ws; (void)ws_size;

    dim3 grid(B_ * NTILES);   // 128 workgroups
    dim3 block(THREADS);      // 128 threads = 4 wave32
    convdeepset_kernel<<<grid, block, 0, stream>>>(x, y, t, sigma, w, bias, out);
}